// ZT_ORAM_Emb_38405597561599
// MI455X (gfx1250) — compile-verified
//
#include <hip/hip_runtime.h>

// Embedding row-gather: out[row] = weight[x[row]] for 819,200 rows of 64 fp32.
// Memory-bound: ~420 MB moved per launch -> ~18 us floor at 23.3 TB/s HBM.
// Strategy: 16 lanes per row, one float4 (b128) per lane; NT store for the
// streaming output, NT load for read-once indices, RT (default) loads for the
// weight table so repeated indices hit the 192 MB L2.

typedef __attribute__((ext_vector_type(4))) float v4f;

#define VEC_PER_ROW 16  // 64 floats / 4 per float4

__global__ __launch_bounds__(256) void ZT_ORAM_Emb_38405597561599_kernel(
    const int* __restrict__ idx,     // [n_rows] int32 indices into weight
    const v4f* __restrict__ weight,  // [VOCAB * 16] float4 view of [VOCAB,64]
    v4f* __restrict__ out,           // [n_rows * 16] float4 view of output
    int n_rows) {
  const int tid  = blockIdx.x * 256 + (int)threadIdx.x;
  const int row  = tid >> 4;          // which embedding row
  const int lane = tid & 15;          // which float4 within the 64-float row
  if (row >= n_rows) return;

  // Index is read exactly once per row across the whole kernel -> NT load.
  // The 16 lanes of a group request the same dword; coalesced into one L0 req.
  const int w = __builtin_nontemporal_load(idx + row);

  // Weight row read: default (RT) temporal hint -> let hot rows live in L2.
  // 16 lanes x 16 B = one fully coalesced 256 B row read (global_load_b128).
  const v4f v = weight[(size_t)w * VEC_PER_ROW + lane];

  // Output is write-once streaming -> non-temporal store (global_store_b128
  // with NT hint) so we don't evict the weight table from L2.
  __builtin_nontemporal_store(v, out + (size_t)row * VEC_PER_ROW + lane);
}

extern "C" void kernel_launch(void* const* d_in, const int* in_sizes, int n_in,
                              void* d_out, int out_size, void* d_ws, size_t ws_size,
                              hipStream_t stream) {
  (void)n_in; (void)out_size; (void)d_ws; (void)ws_size;
  const int*  idx    = (const int*)d_in[0];   // x: [4096*200] int32
  const v4f*  weight = (const v4f*)d_in[1];   // weight: [1e6, 64] fp32
  v4f*        out    = (v4f*)d_out;           // [4096*200*64] fp32

  const int n_rows = in_sizes[0];                 // 819,200
  const long total_threads = (long)n_rows * VEC_PER_ROW;
  const int  blocks = (int)((total_threads + 255) / 256);

  ZT_ORAM_Emb_38405597561599_kernel<<<blocks, 256, 0, stream>>>(
      idx, weight, out, n_rows);
}